// SbS_55224689492190
// MI455X (gfx1250) — compile-verified
//
#include <hip/hip_runtime.h>
#include <stdint.h>

// Problem dims (fixed by the reference)
#define BB   16
#define SS   128
#define XX   24
#define YY   24
#define NN   64
#define TT   256
#define NPIX (BB * XX * YY)   // 9216

typedef float v2f __attribute__((ext_vector_type(2)));
typedef float v8f __attribute__((ext_vector_type(8)));

// Types for the gfx1250 async-to-LDS builtin (param 0 is int __vector(4) * in
// the global/__device__ address space per the compiler diagnostic).
typedef int i32x4 __attribute__((vector_size(16)));
typedef __attribute__((address_space(1))) i32x4 gl_i32x4;
typedef __attribute__((address_space(3))) i32x4 ls_i32x4;

#ifndef __has_builtin
#define __has_builtin(x) 0
#endif

#if __has_builtin(__builtin_amdgcn_global_load_async_to_lds_b128)
#define USE_ASYNC_LDS 1
#else
#define USE_ASYNC_LDS 0
#endif

// ---------------------------------------------------------------------------
// Kernel A: spike generation.
// One block = 16 pixels, 256 threads (8 waves).
//  - all threads load X[s][p] = input[b,s,x,y] into LDS (128x16 f32 = 8KB)
//  - wave 0 computes the inclusive cumsum over S as a blocked triangular
//    matmul using V_WMMA_F32_16X16X4_F32 (exact f32), writing the CDF back
//    in place
//  - all 256 threads then do 16 binary searches each (16 px * 256 t / 256),
//    emitting spike index (u8) and precombined eps = eps_xy[sp,x,y]*eps_t[t]
// ---------------------------------------------------------------------------
__global__ __launch_bounds__(256) void sbs_spikes_kernel(
    const float* __restrict__ input,
    const float* __restrict__ eps_xy,
    const float* __restrict__ eps_t,
    const float* __restrict__ rv,
    uint8_t* __restrict__ spikes,
    float* __restrict__ eps_out)
{
  __shared__ float cdf[SS * 16];   // X on entry, CDF after wave-0 pass
  const int tid = threadIdx.x;
  const int pix_base = blockIdx.x * 16;

  // ---- load X[s][p] ----
  {
    const int p   = tid & 15;
    const int pix = pix_base + p;
    const int b   = pix / (XX * YY);
    const int rem = pix - b * (XX * YY);
    const int x   = rem / YY;
    const int y   = rem - x * YY;
    const int s0  = tid >> 4;
#pragma unroll
    for (int i = 0; i < 8; ++i) {
      const int s = s0 + i * 16;
      cdf[s * 16 + p] = input[((b * SS + s) * XX + x) * YY + y];
    }
  }
  __syncthreads();

  // ---- wave 0: blocked cumsum over S via f32 WMMA ----
  // C_i = Ltri16 @ X_i + Ones16 @ (sum_{j<i} X_j); the Ones accumulator is
  // itself maintained with chained WMMAs.
  if (tid < 32) {
    const int lane = tid;
    const int half = lane >> 4;   // A/B lane-half selects K offset
    const int col  = lane & 15;
    const int m    = lane & 15;   // A-matrix row for this lane
    v8f acc = {0.f, 0.f, 0.f, 0.f, 0.f, 0.f, 0.f, 0.f};
    v2f a_ones; a_ones.x = 1.0f; a_ones.y = 1.0f;

    for (int blk = 0; blk < 8; ++blk) {
      // B operands: rows of X_blk. VGPR v, lane-half hh -> row k0 + 2*hh + v.
      v2f bb4[4];
#pragma unroll
      for (int ki = 0; ki < 4; ++ki) {
        const int r = blk * 16 + ki * 4 + 2 * half;
        bb4[ki].x = cdf[r * 16 + col];
        bb4[ki].y = cdf[(r + 1) * 16 + col];
      }

      // Triangular part: c = Ltri @ X_blk + acc  (4 chained K=4 WMMAs)
      v8f c = acc;
#pragma unroll
      for (int ki = 0; ki < 4; ++ki) {
        const int kb = ki * 4 + 2 * half;   // K index of a.x for this lane
        v2f a;
        a.x = (kb     <= m) ? 1.0f : 0.0f;
        a.y = (kb + 1 <= m) ? 1.0f : 0.0f;
        c = __builtin_amdgcn_wmma_f32_16x16x4_f32(
                false, a, false, bb4[ki], (short)0, c, false, false);
      }
      // Running column-sum accumulator: acc += Ones @ X_blk
#pragma unroll
      for (int ki = 0; ki < 4; ++ki) {
        acc = __builtin_amdgcn_wmma_f32_16x16x4_f32(
                false, a_ones, false, bb4[ki], (short)0, acc, false, false);
      }
      // Write CDF block in place (rows of X_blk no longer needed).
#pragma unroll
      for (int v = 0; v < 8; ++v) {
        cdf[(blk * 16 + v + 8 * half) * 16 + col] = c[v];
      }
    }
  }
  __syncthreads();

  // ---- binary searches: thread -> (p = tid/16, t = i*16 + tid%16) ----
  {
    const int p   = tid >> 4;
    const int tlo = tid & 15;
    const int pix = pix_base + p;
    const int b   = pix / (XX * YY);
    const int rem = pix - b * (XX * YY);
    const int x   = rem / YY;
    const int y   = rem - x * YY;
    const float total = cdf[(SS - 1) * 16 + p];   // > 0 (inputs positive)

#pragma unroll 4
    for (int i = 0; i < 16; ++i) {
      const int t = i * 16 + tlo;
      // compare unnormalized: cdf[s]/total < r  <=>  cdf[s] < r*total
      const float r = rv[((b * TT + t) * XX + x) * YY + y] * total;
      int lo = 0, hi = SS;
#pragma unroll
      for (int it = 0; it < 7; ++it) {
        const int mid = (lo + hi) >> 1;
        if (cdf[mid * 16 + p] < r) lo = mid + 1; else hi = mid;
      }
      const int sp = (lo > SS - 1) ? (SS - 1) : lo;
      spikes[(size_t)pix * TT + t]  = (uint8_t)sp;
      eps_out[(size_t)pix * TT + t] = eps_xy[(sp * XX + x) * YY + y] * eps_t[t];
    }
  }
}

// ---------------------------------------------------------------------------
// Kernel B: sequential H-dynamics. One wave32 per pixel, 8 pixels per block.
// Weights (128x64 f32 = 32KB) staged to LDS via CDNA5 async-to-LDS path.
// h state: 2 f32 per lane. Per step: uniform LDS gather of weights row,
// elementwise mul, 5-level shfl_xor reduction for <h,w>, normalized update.
// ---------------------------------------------------------------------------
__global__ __launch_bounds__(256) void sbs_dynamics_kernel(
    const float* __restrict__ weights,
    const float* __restrict__ h_init,
    const float* __restrict__ fo_ptr,
    const uint8_t* __restrict__ spikes,
    const float* __restrict__ eps_arr,
    float* __restrict__ out)
{
  __shared__ float w_lds[SS * NN];   // 32KB
  const int tid = threadIdx.x;

#if USE_ASYNC_LDS
  // 2048 float4 chunks / 256 threads = 8 async b128 transfers per thread.
#pragma unroll
  for (int i = 0; i < (SS * NN) / (256 * 4); ++i) {
    const int idx4 = tid + i * 256;
    __builtin_amdgcn_global_load_async_to_lds_b128(
        (gl_i32x4*)(weights + idx4 * 4),
        (ls_i32x4*)(w_lds + idx4 * 4),
        0, 0);
  }
  asm volatile("s_wait_asynccnt 0" ::: "memory");
#else
#pragma unroll
  for (int i = 0; i < (SS * NN) / 256; ++i) {
    const int idx = tid + i * 256;
    w_lds[idx] = weights[idx];
  }
#endif
  __syncthreads();

  const int wave = tid >> 5;
  const int lane = tid & 31;
  const int pix  = blockIdx.x * 8 + wave;
  const int b    = pix / (XX * YY);
  const int rem  = pix - b * (XX * YY);
  const int x    = rem / YY;
  const int y    = rem - x * YY;

  float h0 = h_init[lane];
  float h1 = h_init[lane + 32];
  const float fo    = fo_ptr[0];
  const float f_off = fmaxf(fo, 0.0f);
  const float f_loc = f_off / (float)NN;
  const float c1    = 1.0f + f_off;

  const uint8_t* sp_row = spikes  + (size_t)pix * TT;
  const float*   ep_row = eps_arr + (size_t)pix * TT;

  for (int t0 = 0; t0 < TT; t0 += 32) {
    const int   spv = sp_row[t0 + lane];   // 32 spike indices, one per lane
    const float epv = ep_row[t0 + lane];   // matching eps values
#pragma unroll 8
    for (int i = 0; i < 32; ++i) {
      const int   sp  = __shfl(spv, i, 32);   // wave-uniform broadcast
      const float eps = __shfl(epv, i, 32);
      const float w0  = w_lds[sp * NN + lane];        // conflict-free: 64 banks
      const float w1  = w_lds[sp * NN + 32 + lane];
      const float hw0 = h0 * w0;
      const float hw1 = h1 * w1;
      float d = hw0 + hw1;
      d += __shfl_xor(d, 16, 32);
      d += __shfl_xor(d,  8, 32);
      d += __shfl_xor(d,  4, 32);
      d += __shfl_xor(d,  2, 32);
      d += __shfl_xor(d,  1, 32);
      if (d > 1e-10f) {                       // wave-uniform branch
        const float r   = eps / fmaxf(d, 1e-20f);
        const float inv = 1.0f / (1.0f + eps * c1);
        const float add = eps * f_loc;
        h0 = (h0 + hw0 * r + add) * inv;
        h1 = (h1 + hw1 * r + add) * inv;
      }
    }
  }

  // out[b, n, x, y]
  out[((b * NN + lane) * XX + x) * YY + y]        = h0;
  out[((b * NN + lane + 32) * XX + x) * YY + y]   = h1;
}

// ---------------------------------------------------------------------------
extern "C" void kernel_launch(void* const* d_in, const int* in_sizes, int n_in,
                              void* d_out, int out_size, void* d_ws, size_t ws_size,
                              hipStream_t stream) {
  (void)in_sizes; (void)n_in; (void)out_size; (void)ws_size;
  const float* input   = (const float*)d_in[0];
  const float* eps_xy  = (const float*)d_in[1];
  const float* eps_t   = (const float*)d_in[2];
  const float* weights = (const float*)d_in[3];
  const float* h_init  = (const float*)d_in[4];
  const float* fo      = (const float*)d_in[5];
  const float* rv      = (const float*)d_in[6];
  float* out = (float*)d_out;

  // workspace layout: [ spikes u8 : NPIX*TT ][ eps f32 : NPIX*TT ]
  uint8_t* spikes  = (uint8_t*)d_ws;
  float*   eps_arr = (float*)((char*)d_ws + (size_t)NPIX * TT);

  sbs_spikes_kernel<<<NPIX / 16, 256, 0, stream>>>(input, eps_xy, eps_t, rv,
                                                   spikes, eps_arr);
  sbs_dynamics_kernel<<<NPIX / 8, 256, 0, stream>>>(weights, h_init, fo,
                                                    spikes, eps_arr, out);
}